// AttentionDecoder_61649960567390
// MI455X (gfx1250) — compile-verified
//
#include <hip/hip_runtime.h>
#include <hip/hip_bf16.h>

// ---------------------------------------------------------------------------
// AttentionDecoder for MI455X (gfx1250, wave32)
//
//  * keys = enc @ Wk^T + bk computed once on the WMMA pipe, stored bf16;
//    enc cached bf16 -> 128MB attention working set is L2-resident (192MB L2)
//    for all 500 steps instead of 128GB of HBM reads.
//  * Per-step GEMMs: V_WMMA_F32_16X16X4_F32 (full fp32 for the recurrence),
//    4 M-tiles per wave (one B fragment feeds 4 WMMAs), B128 fragment loads
//    (one load feeds 2 WMMAs via K-slot permutation), distance-1 software
//    prefetch so loads overlap the WMMA stream.
//  * Attention: all keys/enc reads are 128-bit (8 x bf16) with bit-level
//    bf16->f32 conversion; context uses per-thread 8-wide accumulators with
//    a 4-way s-partition reduced in LDS.
// ---------------------------------------------------------------------------

typedef __attribute__((ext_vector_type(2))) float v2f;
typedef __attribute__((ext_vector_type(4))) float v4f;
typedef __attribute__((ext_vector_type(8))) float v8f;

#define BATCH  64
#define SEQ    1024
#define HID    512
#define NMELS  80
#define GATES  2048
#define XDIM   1536   // 2H + E

static __device__ __forceinline__ void bf2x(unsigned u, float& f0, float& f1)
{
    f0 = __uint_as_float(u << 16);
    f1 = __uint_as_float(u & 0xffff0000u);
}

static __device__ __forceinline__ v8f wmma4(v2f a, v2f b, v8f c)
{
    // 8 args: (neg_a, A, neg_b, B, c_mod, C, reuse_a, reuse_b)
    return __builtin_amdgcn_wmma_f32_16x16x4_f32(false, a, false, b,
                                                 (short)0, c, false, false);
}

// ---------------------------------------------------------------------------
// Row-major GEMM on the WMMA pipe:
//   out[M x N] = act( A[M x K] @ W[N x K]^T + bias[N] + (ACCUM ? out : 0) )
//
// One wave computes MT vertically-stacked 16x16 output tiles; the B fragment
// is loaded once per K-chunk and reused by MT WMMAs.
//
// K-slot permutation for B128 loads: each lane loads 4 contiguous floats at
// k0 + 4*(lane>>4). WMMA reduces over physical slots, so as long as A and B
// use the same lane->k mapping the result is exact:
//   floats [0,1] -> chunk {k0,k0+1, k0+4,k0+5}
//   floats [2,3] -> chunk {k0+2,k0+3, k0+6,k0+7}
// Two WMMAs per 128-bit load per operand; K advances by 8.
//
// C/D layout per ISA 7.12.2: VGPR r -> M = r + 8*(lane>>4), N = lane&15.
// Requires M % (16*MT) == 0, N % 16 == 0, K % 8 == 0 (true for all shapes).
// ---------------------------------------------------------------------------
template <typename OutT, bool RELU, bool ACCUM, int MT>
__global__ void wmma_gemm_f32(const float* __restrict__ A,
                              const float* __restrict__ W,
                              const float* __restrict__ bias,
                              OutT* __restrict__ out,
                              int M, int N, int K)
{
    const int lane    = threadIdx.x & 31;
    const int waveId  = (int)((blockIdx.x * blockDim.x + threadIdx.x) >> 5);
    const int ntN     = N >> 4;
    const int nGroups = ((M >> 4) / MT) * ntN;
    if (waveId >= nGroups) return;        // wave-uniform: EXEC all-1 below

    const int gM = waveId / ntN;          // group of MT M-tiles
    const int tN = waveId - gM * ntN;

    const int kk4  = (lane >> 4) * 4;     // {0,4}: contiguous K span per lane
    const int bcol = tN * 16 + (lane & 15);
    const int ccol = bcol;

    const float* aBase[MT];
#pragma unroll
    for (int m = 0; m < MT; ++m)
        aBase[m] = A + (size_t)(gM * MT * 16 + m * 16 + (lane & 15)) * K + kk4;
    const float* bBase = W + (size_t)bcol * K + kk4;

    v8f acc[MT];
#pragma unroll
    for (int m = 0; m < MT; ++m) acc[m] = (v8f){};
    if (ACCUM) {
#pragma unroll
        for (int m = 0; m < MT; ++m) {
            const int crow0 = gM * MT * 16 + m * 16 + (lane >> 4) * 8;
#pragma unroll
            for (int r = 0; r < 8; ++r)
                acc[m][r] = (float)out[(size_t)(crow0 + r) * N + ccol];
        }
    }

    // ---- software-pipelined K loop (distance-1 prefetch) -----------------
    v4f bq = *(const v4f*)bBase;
    v4f aq[MT];
#pragma unroll
    for (int m = 0; m < MT; ++m) aq[m] = *(const v4f*)aBase[m];

    for (int k0 = 0; k0 + 8 < K; k0 += 8) {
        v4f bn = *(const v4f*)(bBase + k0 + 8);
        v4f an[MT];
#pragma unroll
        for (int m = 0; m < MT; ++m) an[m] = *(const v4f*)(aBase[m] + k0 + 8);

        const v2f blo = __builtin_shufflevector(bq, bq, 0, 1);
        const v2f bhi = __builtin_shufflevector(bq, bq, 2, 3);
#pragma unroll
        for (int m = 0; m < MT; ++m) {
            acc[m] = wmma4(__builtin_shufflevector(aq[m], aq[m], 0, 1), blo, acc[m]);
            acc[m] = wmma4(__builtin_shufflevector(aq[m], aq[m], 2, 3), bhi, acc[m]);
        }
        bq = bn;
#pragma unroll
        for (int m = 0; m < MT; ++m) aq[m] = an[m];
    }
    {   // epilogue: last K pair
        const v2f blo = __builtin_shufflevector(bq, bq, 0, 1);
        const v2f bhi = __builtin_shufflevector(bq, bq, 2, 3);
#pragma unroll
        for (int m = 0; m < MT; ++m) {
            acc[m] = wmma4(__builtin_shufflevector(aq[m], aq[m], 0, 1), blo, acc[m]);
            acc[m] = wmma4(__builtin_shufflevector(aq[m], aq[m], 2, 3), bhi, acc[m]);
        }
    }

    const float bv = bias ? bias[ccol] : 0.0f;
#pragma unroll
    for (int m = 0; m < MT; ++m) {
        const int crow0 = gM * MT * 16 + m * 16 + (lane >> 4) * 8;
#pragma unroll
        for (int r = 0; r < 8; ++r) {
            float v = acc[m][r] + bv;
            if (RELU) v = fmaxf(v, 0.0f);
            out[(size_t)(crow0 + r) * N + ccol] = (OutT)v;
        }
    }
}

// ---------------------------------------------------------------------------
__global__ void f32_to_bf16(const float* __restrict__ in,
                            __hip_bfloat16* __restrict__ out, size_t n)
{
    size_t i = (size_t)blockIdx.x * blockDim.x + threadIdx.x;
    if (i < n) out[i] = __float2bfloat16(in[i]);
}

__global__ void zero_f32(float* __restrict__ p, int n)
{
    int i = blockIdx.x * blockDim.x + threadIdx.x;
    if (i < n) p[i] = 0.0f;
}

// ---------------------------------------------------------------------------
// Attention: one block (256 threads) per batch row.
//   scores = q . keys   (B128 bf16 loads, bit-converted, fp32 FMA)
//   softmax (block reduction)
//   ctx    = softmax(scores) . enc  (B128 bf16 loads, 8-wide accumulators,
//                                    4-way s-partition reduced in LDS)
// ---------------------------------------------------------------------------
__global__ void attention_kernel(const float* __restrict__ q,             // B x H
                                 const __hip_bfloat16* __restrict__ keys, // B x S x H
                                 const __hip_bfloat16* __restrict__ enc,  // B x S x H
                                 float* __restrict__ ctx)                 // B x H
{
    __shared__ float sQ[HID];
    __shared__ float sS[SEQ];
    __shared__ float red[8];
    __shared__ float pctx[4][HID];

    const int b    = blockIdx.x;
    const int tid  = threadIdx.x;      // 0..255
    const int lane = tid & 31;
    const int wv   = tid >> 5;

    sQ[tid]       = q[b * HID + tid];
    sQ[tid + 256] = q[b * HID + tid + 256];
    __syncthreads();

    // ---- scores -----------------------------------------------------------
    const __hip_bfloat16* kb = keys + (size_t)b * SEQ * HID;
    float lmax = -3.4e38f;
    for (int s = tid; s < SEQ; s += 256) {
        const uint4* row = (const uint4*)(kb + (size_t)s * HID);
        float acc = 0.0f;
#pragma unroll 4
        for (int j = 0; j < HID / 8; ++j) {      // 64 x 128-bit loads
            uint4 u = row[j];
            const int k0 = j * 8;
            float f0, f1;
            bf2x(u.x, f0, f1); acc += sQ[k0 + 0] * f0 + sQ[k0 + 1] * f1;
            bf2x(u.y, f0, f1); acc += sQ[k0 + 2] * f0 + sQ[k0 + 3] * f1;
            bf2x(u.z, f0, f1); acc += sQ[k0 + 4] * f0 + sQ[k0 + 5] * f1;
            bf2x(u.w, f0, f1); acc += sQ[k0 + 6] * f0 + sQ[k0 + 7] * f1;
        }
        sS[s] = acc;
        lmax = fmaxf(lmax, acc);
    }

    // ---- softmax ----------------------------------------------------------
    for (int off = 16; off; off >>= 1)
        lmax = fmaxf(lmax, __shfl_xor(lmax, off, 32));
    if (lane == 0) red[wv] = lmax;
    __syncthreads();
    float gmax = red[0];
#pragma unroll
    for (int i = 1; i < 8; ++i) gmax = fmaxf(gmax, red[i]);

    float lsum = 0.0f;
    for (int s = tid; s < SEQ; s += 256) {
        float e = __expf(sS[s] - gmax);
        sS[s] = e;
        lsum += e;
    }
    for (int off = 16; off; off >>= 1) lsum += __shfl_xor(lsum, off, 32);
    __syncthreads();                       // gmax reads done, sS fully written
    if (lane == 0) red[wv] = lsum;
    __syncthreads();
    float gsum = 0.0f;
#pragma unroll
    for (int i = 0; i < 8; ++i) gsum += red[i];
    const float inv = 1.0f / gsum;

    // ---- context ----------------------------------------------------------
    // thread -> 8 contiguous h; 4 groups partition s; reduce partials in LDS.
    const int g  = tid >> 6;               // 0..3
    const int h0 = (tid & 63) * 8;
    const __hip_bfloat16* eb = enc + (size_t)b * SEQ * HID;

    float a8[8];
#pragma unroll
    for (int j = 0; j < 8; ++j) a8[j] = 0.0f;

    const int sEnd = g * 256 + 256;
    for (int s = g * 256; s < sEnd; ++s) {
        const float w = sS[s];
        uint4 u = *(const uint4*)(eb + (size_t)s * HID + h0);
        float f0, f1;
        bf2x(u.x, f0, f1); a8[0] += w * f0; a8[1] += w * f1;
        bf2x(u.y, f0, f1); a8[2] += w * f0; a8[3] += w * f1;
        bf2x(u.z, f0, f1); a8[4] += w * f0; a8[5] += w * f1;
        bf2x(u.w, f0, f1); a8[6] += w * f0; a8[7] += w * f1;
    }
#pragma unroll
    for (int j = 0; j < 8; ++j) pctx[g][h0 + j] = a8[j];
    __syncthreads();

    for (int hh = tid; hh < HID; hh += 256)
        ctx[b * HID + hh] =
            (pctx[0][hh] + pctx[1][hh] + pctx[2][hh] + pctx[3][hh]) * inv;
}

// ---------------------------------------------------------------------------
__global__ void build_x(const float* __restrict__ p,
                        const float* __restrict__ ctx,
                        const float* __restrict__ spk,
                        float* __restrict__ x)
{
    int i = blockIdx.x * blockDim.x + threadIdx.x;   // BATCH * XDIM
    if (i >= BATCH * XDIM) return;
    int b = i / XDIM, j = i - b * XDIM;
    float v;
    if (j < HID)            v = p[b * HID + j];
    else if (j < 2 * HID)   v = ctx[b * HID + (j - HID)];
    else                    v = spk[b * HID + (j - 2 * HID)];
    x[i] = v;
}

__global__ void lstm_update(const float* __restrict__ gates,   // B x 4H (i,f,g,o)
                            float* __restrict__ h, float* __restrict__ c)
{
    int i = blockIdx.x * blockDim.x + threadIdx.x;   // BATCH * HID
    if (i >= BATCH * HID) return;
    int b = i >> 9, k = i & (HID - 1);
    const float* g = gates + (size_t)b * GATES;
    float ig = g[k], fg = g[HID + k], gg = g[2 * HID + k], og = g[3 * HID + k];
    float si = 1.0f / (1.0f + __expf(-ig));
    float sf = 1.0f / (1.0f + __expf(-fg));
    float so = 1.0f / (1.0f + __expf(-og));
    float cn = sf * c[i] + si * tanhf(gg);
    float hn = so * tanhf(cn);
    c[i] = cn;
    h[i] = hn;
}

// mel copy into (B, T, 80) output + stop projection into (B, T)
__global__ void emit_kernel(const float* __restrict__ mel,  // B x 80
                            const float* __restrict__ h,    // B x H
                            const float* __restrict__ Ws,   // 1 x H
                            const float* __restrict__ bs,   // 1
                            float* __restrict__ outMel,
                            float* __restrict__ outStop,
                            int t, int T)
{
    __shared__ float red[128];
    const int b = blockIdx.x;
    const int tid = threadIdx.x;                      // 128 threads
    if (tid < NMELS)
        outMel[((size_t)b * T + t) * NMELS + tid] = mel[b * NMELS + tid];
    float s = 0.0f;
    for (int k = tid; k < HID; k += 128) s += h[b * HID + k] * Ws[k];
    red[tid] = s;
    __syncthreads();
    for (int off = 64; off; off >>= 1) {
        if (tid < off) red[tid] += red[tid + off];
        __syncthreads();
    }
    if (tid == 0) outStop[(size_t)b * T + t] = red[0] + bs[0];
}

// ---------------------------------------------------------------------------
static inline int gemm_blocks(int M, int N, int MT)
{
    int waves = ((M / 16) / MT) * (N / 16);
    return (waves * 32 + 255) / 256;
}

extern "C" void kernel_launch(void* const* d_in, const int* in_sizes, int n_in,
                              void* d_out, int out_size, void* d_ws, size_t ws_size,
                              hipStream_t stream)
{
    const float* enc  = (const float*)d_in[0];   // (64,1024,512)
    const float* spk  = (const float*)d_in[1];   // (64,512)
    // d_in[2] = max_steps on device; derive T host-side from out_size instead.
    const float* W1   = (const float*)d_in[3];
    const float* b1   = (const float*)d_in[4];
    const float* W2   = (const float*)d_in[5];
    const float* b2   = (const float*)d_in[6];
    const float* Wq   = (const float*)d_in[7];
    const float* bq   = (const float*)d_in[8];
    const float* Wk   = (const float*)d_in[9];
    const float* bk   = (const float*)d_in[10];
    const float* W_ih = (const float*)d_in[11];
    const float* b_ih = (const float*)d_in[12];
    const float* W_hh = (const float*)d_in[13];
    const float* b_hh = (const float*)d_in[14];
    const float* Wm   = (const float*)d_in[15];
    const float* bm   = (const float*)d_in[16];
    const float* Ws   = (const float*)d_in[17];
    const float* bs   = (const float*)d_in[18];

    const int T = out_size / (BATCH * (NMELS + 1));  // == max_steps (500)

    // ---- workspace layout (needs ~136 MB) --------------------------------
    char* ws = (char*)d_ws;
    const size_t bfBytes = (size_t)BATCH * SEQ * HID * sizeof(__hip_bfloat16); // 64MB
    __hip_bfloat16* keys_bf = (__hip_bfloat16*)ws;
    __hip_bfloat16* enc_bf  = (__hip_bfloat16*)(ws + bfBytes);
    float* f     = (float*)(ws + 2 * bfBytes);
    float* out1  = f;                 f += BATCH * HID;
    float* p     = f;                 f += BATCH * HID;
    float* q     = f;                 f += BATCH * HID;
    float* ctx   = f;                 f += BATCH * HID;
    float* x     = f;                 f += BATCH * XDIM;
    float* gates = f;                 f += BATCH * GATES;
    float* h     = f;                 f += BATCH * HID;
    float* c     = f;                 f += BATCH * HID;
    float* mel   = f;                 f += BATCH * NMELS;

    float* outMel  = (float*)d_out;
    float* outStop = outMel + (size_t)BATCH * T * NMELS;

    // ---- one-time: bf16 enc cache + keys = enc @ Wk^T + bk (bf16) --------
    const size_t nEnc = (size_t)BATCH * SEQ * HID;
    f32_to_bf16<<<(int)((nEnc + 255) / 256), 256, 0, stream>>>(enc, enc_bf, nEnc);
    wmma_gemm_f32<__hip_bfloat16, false, false, 4>
        <<<gemm_blocks(BATCH * SEQ, HID, 4), 256, 0, stream>>>(
            enc, Wk, bk, keys_bf, BATCH * SEQ, HID, HID);

    // ---- deterministic state init ----------------------------------------
    zero_f32<<<(BATCH * HID + 255) / 256, 256, 0, stream>>>(h, BATCH * HID);
    zero_f32<<<(BATCH * HID + 255) / 256, 256, 0, stream>>>(c, BATCH * HID);
    zero_f32<<<(BATCH * NMELS + 255) / 256, 256, 0, stream>>>(mel, BATCH * NMELS);

    // ---- autoregressive loop ---------------------------------------------
    for (int t = 0; t < T; ++t) {
        // prenet
        wmma_gemm_f32<float, true, false, 4>
            <<<gemm_blocks(BATCH, HID, 4), 256, 0, stream>>>(
                mel, W1, b1, out1, BATCH, HID, NMELS);
        wmma_gemm_f32<float, true, false, 4>
            <<<gemm_blocks(BATCH, HID, 4), 256, 0, stream>>>(
                out1, W2, b2, p, BATCH, HID, HID);
        // query (uses pre-update h)
        wmma_gemm_f32<float, false, false, 4>
            <<<gemm_blocks(BATCH, HID, 4), 256, 0, stream>>>(
                h, Wq, bq, q, BATCH, HID, HID);
        // attention over L2-resident bf16 keys/enc
        attention_kernel<<<BATCH, 256, 0, stream>>>(q, keys_bf, enc_bf, ctx);
        // x = [p, ctx, spk]
        build_x<<<(BATCH * XDIM + 255) / 256, 256, 0, stream>>>(p, ctx, spk, x);
        // gates = x @ W_ih^T + b_ih + h @ W_hh^T + b_hh
        wmma_gemm_f32<float, false, false, 4>
            <<<gemm_blocks(BATCH, GATES, 4), 256, 0, stream>>>(
                x, W_ih, b_ih, gates, BATCH, GATES, XDIM);
        wmma_gemm_f32<float, false, true, 4>
            <<<gemm_blocks(BATCH, GATES, 4), 256, 0, stream>>>(
                h, W_hh, b_hh, gates, BATCH, GATES, HID);
        // LSTM cell update
        lstm_update<<<(BATCH * HID + 255) / 256, 256, 0, stream>>>(gates, h, c);
        // mel projection (N = 80 = 5 x 16 tiles)
        wmma_gemm_f32<float, false, false, 4>
            <<<gemm_blocks(BATCH, NMELS, 4), 256, 0, stream>>>(
                h, Wm, bm, mel, BATCH, NMELS, HID);
        // emit mel row into (B,T,80) layout + stop token
        emit_kernel<<<BATCH, 128, 0, stream>>>(mel, h, Ws, bs,
                                               outMel, outStop, t, T);
    }
}